// SparseBox3DDecoderLite_16097537425504
// MI455X (gfx1250) — compile-verified
//
#include <hip/hip_runtime.h>
#include <math.h>

#ifndef INFINITY
#define INFINITY __builtin_inff()
#endif

static constexpr int BB   = 8;        // batches
static constexpr int QQ   = 100000;   // queries
static constexpr int CC   = 10;       // classes
static constexpr int QC   = QQ * CC;  // 1,000,000
static constexpr int KK   = 1000;     // max detections
static constexpr int BINS = 4096;     // 12-bit ordered-key histogram
static constexpr int CAND = 8192;     // candidate cap per batch (bitonic size)

// Monotonic float -> uint mapping (ascending float == ascending uint)
__device__ __forceinline__ unsigned ordKey(float f) {
    unsigned b = __float_as_uint(f);
    return (b & 0x80000000u) ? ~b : (b ^ 0x80000000u);
}
__device__ __forceinline__ float ordInv(unsigned u) {
    unsigned b = (u & 0x80000000u) ? (u ^ 0x80000000u) : ~u;
    return __uint_as_float(b);
}
__device__ __forceinline__ float sigm(float x) { return 1.0f / (1.0f + expf(-x)); }

// ---------------------------------------------------------------------------
// Kernel 1: per-batch 4096-bin histogram of ordered keys (LDS-accumulated)
// ---------------------------------------------------------------------------
__global__ void k_hist(const float* __restrict__ cls, unsigned* __restrict__ hist) {
    __shared__ unsigned lh[BINS];
    const int b = blockIdx.y;
    for (int i = threadIdx.x; i < BINS; i += blockDim.x) lh[i] = 0u;
    __syncthreads();

    const float4* p = reinterpret_cast<const float4*>(cls + (size_t)b * QC);
    const int n4 = QC / 4;
    const int stride = gridDim.x * blockDim.x;
    for (int i = blockIdx.x * blockDim.x + threadIdx.x; i < n4; i += stride) {
        __builtin_prefetch(&p[i + stride], 0, 0);   // global_prefetch_b8, speculative
        float4 v = p[i];
        atomicAdd(&lh[ordKey(v.x) >> 20], 1u);
        atomicAdd(&lh[ordKey(v.y) >> 20], 1u);
        atomicAdd(&lh[ordKey(v.z) >> 20], 1u);
        atomicAdd(&lh[ordKey(v.w) >> 20], 1u);
    }
    __syncthreads();
    unsigned* gh = hist + (size_t)b * BINS;
    for (int i = threadIdx.x; i < BINS; i += blockDim.x) {
        unsigned c = lh[i];
        if (c) atomicAdd(&gh[i], c);
    }
}

// ---------------------------------------------------------------------------
// Kernel 2: find the key threshold (bin) such that count(keys >= thr) >= K
// ---------------------------------------------------------------------------
__global__ void k_thresh(const unsigned* __restrict__ hist, unsigned* __restrict__ thr) {
    const int b = blockIdx.x;
    if (threadIdx.x == 0) {
        const unsigned* h = hist + (size_t)b * BINS;
        unsigned cum = 0;
        int bin = BINS - 1;
        for (; bin >= 0; --bin) {
            cum += h[bin];
            if (cum >= (unsigned)KK) break;
        }
        if (bin < 0) bin = 0;
        thr[b] = ((unsigned)bin) << 20;
    }
}

// ---------------------------------------------------------------------------
// Kernel 3: compact candidates >= threshold, wave32 ballot-aggregated atomics
// ---------------------------------------------------------------------------
__global__ void k_compact(const float* __restrict__ cls,
                          const unsigned* __restrict__ thr,
                          unsigned* __restrict__ cnt,
                          unsigned* __restrict__ candK,
                          unsigned* __restrict__ candI) {
    const int b = blockIdx.y;
    const unsigned ut = thr[b];
    const float* p = cls + (size_t)b * QC;
    const int stride = gridDim.x * blockDim.x;
    unsigned* ck = candK + (size_t)b * CAND;
    unsigned* ci = candI + (size_t)b * CAND;

    for (int i = blockIdx.x * blockDim.x + threadIdx.x; i < QC; i += stride) {
        __builtin_prefetch(&p[i + stride], 0, 0);
        unsigned u = ordKey(p[i]);
        bool pred = (u >= ut);
        unsigned mask = (unsigned)__ballot(pred);   // wave32: low 32 bits
        if (mask) {
            int lane = threadIdx.x & 31;
            int leader = __ffs(mask) - 1;
            unsigned base = 0;
            if (lane == leader) base = atomicAdd(&cnt[b], (unsigned)__popc(mask));
            base = (unsigned)__shfl((int)base, leader, 32);
            if (pred) {
                unsigned pos = base + __popc(mask & ((1u << lane) - 1u));
                if (pos < (unsigned)CAND) { ck[pos] = u; ci[pos] = (unsigned)i; }
            }
        }
    }
}

// ---------------------------------------------------------------------------
// Kernel 4: one block per batch. Async-DMA candidates into LDS
// (GLOBAL_LOAD_ASYNC_TO_LDS_B128, ASYNCcnt), bitonic-sort 8192 by raw key,
// take top K, rerank with sigmoid(centerness), bitonic-resort 1024, decode.
// ---------------------------------------------------------------------------
__global__ void __launch_bounds__(1024)
k_finalize(const unsigned* __restrict__ candK,
           const unsigned* __restrict__ candI,
           const unsigned* __restrict__ cnt,
           const float* __restrict__ quality,
           const float* __restrict__ boxp,
           float* __restrict__ out, int labelsI64) {
    __shared__ unsigned sK[CAND];
    __shared__ unsigned sV[CAND];

    const int b = blockIdx.x;
    const unsigned n = min(cnt[b], (unsigned)CAND);
    const unsigned* ck = candK + (size_t)b * CAND;
    const unsigned* ci = candI + (size_t)b * CAND;

    // ---- async copy of both 32KB candidate arrays straight into LDS ------
    // the instruction offset is added to both the LDS and the global address
    {
        const unsigned ldsK = (unsigned)(size_t)&sK[0];   // low 32 bits = LDS byte addr
        const unsigned ldsV = (unsigned)(size_t)&sV[0];
        const unsigned t16  = (unsigned)threadIdx.x * 16u;  // 1024 thr * 16B = 16KB
        asm volatile("global_load_async_to_lds_b128 %0, %1, %2 offset:0"
                     :: "v"(ldsK + t16), "v"(t16), "s"(ck) : "memory");
        asm volatile("global_load_async_to_lds_b128 %0, %1, %2 offset:16384"
                     :: "v"(ldsK + t16), "v"(t16), "s"(ck) : "memory");
        asm volatile("global_load_async_to_lds_b128 %0, %1, %2 offset:0"
                     :: "v"(ldsV + t16), "v"(t16), "s"(ci) : "memory");
        asm volatile("global_load_async_to_lds_b128 %0, %1, %2 offset:16384"
                     :: "v"(ldsV + t16), "v"(t16), "s"(ci) : "memory");
        asm volatile("s_wait_asynccnt 0x0" ::: "memory");
    }
    __syncthreads();
    // zero the stale tail beyond the live candidate count
    for (unsigned i = n + threadIdx.x; i < (unsigned)CAND; i += blockDim.x) {
        sK[i] = 0u; sV[i] = 0u;
    }
    __syncthreads();

    // ---- bitonic sort CAND entries, descending by key --------------------
    for (unsigned k = 2; k <= (unsigned)CAND; k <<= 1) {
        for (unsigned j = k >> 1; j > 0; j >>= 1) {
            for (unsigned i = threadIdx.x; i < (unsigned)CAND; i += blockDim.x) {
                unsigned ixj = i ^ j;
                if (ixj > i) {
                    unsigned a = sK[i], c = sK[ixj];
                    bool up = ((i & k) == 0);
                    bool sw = up ? (a < c) : (a > c);
                    if (sw) {
                        sK[i] = c; sK[ixj] = a;
                        unsigned t = sV[i]; sV[i] = sV[ixj]; sV[ixj] = t;
                    }
                }
            }
            __syncthreads();
        }
    }

    // ---- pull top 1024 into registers, then reuse LDS for stage 2 --------
    const int t = threadIdx.x;                 // blockDim.x == 1024
    unsigned myKey = sK[t];
    unsigned myIdx = sV[t];
    __syncthreads();

    float*    rr   = reinterpret_cast<float*>(sK);        // [0..1023] rerank key
    unsigned* slot = sV;                                  // [0..1023] payload slot
    unsigned* qA   = sK + 1024;                           // query index
    unsigned* cA   = sK + 2048;                           // class id
    float*    rawA = reinterpret_cast<float*>(sK + 3072); // raw sigmoid score

    bool valid = (t < KK) && ((unsigned)t < n);
    float logit = ordInv(myKey);
    unsigned q  = myIdx / (unsigned)CC;
    unsigned cid = myIdx % (unsigned)CC;
    float s = sigm(logit);
    float cen = valid ? quality[((size_t)b * QQ + q) * 2] : 0.0f;
    float r = valid ? s * sigm(cen) : -INFINITY;

    rr[t] = r; slot[t] = (unsigned)t; qA[t] = q; cA[t] = cid; rawA[t] = valid ? s : 0.0f;
    __syncthreads();

    // ---- bitonic sort 1024 (rr, slot), descending ------------------------
    for (unsigned k = 2; k <= 1024u; k <<= 1) {
        for (unsigned j = k >> 1; j > 0; j >>= 1) {
            unsigned i = (unsigned)t;
            unsigned ixj = i ^ j;
            if (ixj > i) {
                float a = rr[i], c = rr[ixj];
                bool up = ((i & k) == 0);
                bool sw = up ? (a < c) : (a > c);
                if (sw) {
                    rr[i] = c; rr[ixj] = a;
                    unsigned tv = slot[i]; slot[i] = slot[ixj]; slot[ixj] = tv;
                }
            }
            __syncthreads();
        }
    }

    // ---- outputs ---------------------------------------------------------
    if (t < KK) {
        float rv = rr[t];
        unsigned sl = slot[t];
        unsigned qi = qA[sl];
        unsigned li = cA[sl];
        float raw = rawA[sl];

        const float* bp = boxp + ((size_t)b * QQ + qi) * 10;
        float bx = bp[0], by = bp[1], bz = bp[2];
        float w = expf(bp[3]), l = expf(bp[4]), h = expf(bp[5]);
        float yaw = atan2f(bp[6], bp[7]);
        float vx = bp[8], vy = bp[9];

        float* boxes  = out;                         // [B,K,10]
        float* scores = out + (size_t)BB * KK * 10;  // [B,K]
        float* ob = boxes + ((size_t)b * KK + t) * 10;
        ob[0] = bx; ob[1] = by; ob[2] = bz; ob[3] = w; ob[4] = l; ob[5] = h;
        ob[6] = yaw; ob[7] = vx; ob[8] = vy; ob[9] = 0.0f;
        scores[(size_t)b * KK + t] = rv;

        if (labelsI64) {
            long long* labels = reinterpret_cast<long long*>(scores + (size_t)BB * KK);
            float* raws = reinterpret_cast<float*>(labels + (size_t)BB * KK);
            labels[(size_t)b * KK + t] = (long long)li;
            raws[(size_t)b * KK + t] = raw;
        } else {
            float* labels = scores + (size_t)BB * KK;
            float* raws = labels + (size_t)BB * KK;
            labels[(size_t)b * KK + t] = (float)li;
            raws[(size_t)b * KK + t] = raw;
        }
    }
}

// ---------------------------------------------------------------------------
extern "C" void kernel_launch(void* const* d_in, const int* in_sizes, int n_in,
                              void* d_out, int out_size, void* d_ws, size_t ws_size,
                              hipStream_t stream) {
    const float* cls  = (const float*)d_in[0];   // [8,100000,10]
    const float* boxp = (const float*)d_in[1];   // [8,100000,10]
    const float* qual = (const float*)d_in[2];   // [8,100000,2]
    float* out = (float*)d_out;

    unsigned* W    = (unsigned*)d_ws;
    unsigned* HIST = W;                          // 8*4096
    unsigned* CNT  = W + BB * BINS;              // 8
    unsigned* THR  = CNT + BB;                   // 8
    unsigned* CK   = THR + BB;                   // 8*8192
    unsigned* CI   = CK + (size_t)BB * CAND;     // 8*8192

    // zero histograms + counters (thresholds overwritten by k_thresh)
    hipMemsetAsync(W, 0, (size_t)(BB * BINS + 2 * BB) * sizeof(unsigned), stream);

    dim3 gs(128, BB), bs(256);
    k_hist<<<gs, bs, 0, stream>>>(cls, HIST);
    k_thresh<<<BB, 32, 0, stream>>>(HIST, THR);
    k_compact<<<gs, bs, 0, stream>>>(cls, THR, CNT, CK, CI);

    // labels_3d is int64 in the reference; detect how the harness counted it.
    const int sz_i64 = BB * KK * 10 + BB * KK + 2 * BB * KK + BB * KK; // 112000
    int labelsI64 = (out_size == sz_i64) ? 1 : 0;

    k_finalize<<<BB, 1024, 0, stream>>>(CK, CI, CNT, qual, boxp, out, labelsI64);
}